// Stage2Loss_75737453298215
// MI455X (gfx1250) — compile-verified
//
#include <hip/hip_runtime.h>
#include <math.h>

typedef float v2f __attribute__((ext_vector_type(2)));
typedef float v8f __attribute__((ext_vector_type(8)));

#define NUM_IDS 25
#define N_BT 8
#define HPIX 64
#define WPIX 64
#define PIX_SCENE 4096
#define C_SEM 512
#define C_INS 64
#define HH 128
#define WW 128
#define TILE_PIX 256
#define TILES_SCENE 16

#define PS_STRIDE 513   // 512 + 1 pad -> bank = (m+c)%64
#define PI_STRIDE 65    // 64 + 1 pad

// workspace layout (float offsets)
#define OFF_PS_SUM  0                         // 200*512 = 102400  (sums, then normalized in place)
#define OFF_PI_SUM  102400                    // 200*64  = 12800
#define OFF_CNT     115200                    // 200
#define OFF_ACC     115400                    // 8: [sa_sum, valid_cnt, sm_sum, mask_sum, ii_sum, im_sum, inter_sum, pair_cnt]
#define OFF_INV_GT  115408                    // 32768
#define OFF_INV_INS 148176                    // 32768
#define WS_FLOATS   180944

__device__ __forceinline__ float wave_sum(float v) {
  for (int off = 16; off > 0; off >>= 1) v += __shfl_xor(v, off, 32);
  return v;
}

// ---------------- K1a: per-pixel inverse norms + per-segment counts -------------
__global__ void k1a_norms_counts(const float* __restrict__ lseg_gt,
                                 const float* __restrict__ rins,
                                 const int*   __restrict__ imask,
                                 float* __restrict__ ws) {
  __shared__ float cnt_s[NUM_IDS];
  const int tid  = threadIdx.x;
  const int bt   = blockIdx.x / TILES_SCENE;
  const int tile = blockIdx.x % TILES_SCENE;
  if (tid < NUM_IDS) cnt_s[tid] = 0.0f;
  __syncthreads();

  const int p  = tile * TILE_PIX + tid;          // 0..4095
  const int hp = p >> 6, wp = p & 63;
  const int m  = imask[(size_t)bt * HH * WW + (2 * hp) * WW + 2 * wp];

  // sumsq of 2x2-mean-downsampled gt over 512 channels (coalesced v2f loads)
  const float* gt0 = lseg_gt + (size_t)bt * C_SEM * HH * WW + (2 * hp) * WW + 2 * wp;
  float ss = 0.0f;
  for (int c = 0; c < C_SEM; ++c) {
    const float* pr = gt0 + (size_t)c * HH * WW;
    v2f r0 = *(const v2f*)pr;
    v2f r1 = *(const v2f*)(pr + WW);
    float v = 0.25f * (r0.x + r0.y + r1.x + r1.y);
    ss += v * v;
  }
  ws[OFF_INV_GT + bt * PIX_SCENE + p] = 1.0f / fmaxf(sqrtf(ss), 1e-12f);

  const float* ri0 = rins + (size_t)bt * C_INS * PIX_SCENE + p;
  float si = 0.0f;
  for (int c = 0; c < C_INS; ++c) { float v = ri0[c * PIX_SCENE]; si += v * v; }
  ws[OFF_INV_INS + bt * PIX_SCENE + p] = 1.0f / fmaxf(sqrtf(si), 1e-12f);

  atomicAdd(&cnt_s[m], 1.0f);
  __syncthreads();
  if (tid < NUM_IDS) atomicAdd(&ws[OFF_CNT + bt * NUM_IDS + tid], cnt_s[tid]);
}

// ---------------- K1b: accumulate normalized vectors into prototype sums --------
__global__ void k1b_proto_accum(const float* __restrict__ lseg_gt,
                                const float* __restrict__ rins,
                                const int*   __restrict__ imask,
                                float* __restrict__ ws) {
  __shared__ float ps[NUM_IDS * PS_STRIDE];   // ~51.3 KB
  __shared__ float pi[NUM_IDS * PI_STRIDE];
  const int tid  = threadIdx.x;
  const int bt   = blockIdx.x / TILES_SCENE;
  const int tile = blockIdx.x % TILES_SCENE;
  for (int i = tid; i < NUM_IDS * PS_STRIDE; i += blockDim.x) ps[i] = 0.0f;
  for (int i = tid; i < NUM_IDS * PI_STRIDE; i += blockDim.x) pi[i] = 0.0f;
  __syncthreads();

  const int p  = tile * TILE_PIX + tid;
  const int hp = p >> 6, wp = p & 63;
  const int m  = imask[(size_t)bt * HH * WW + (2 * hp) * WW + 2 * wp];
  const float invg = ws[OFF_INV_GT + bt * PIX_SCENE + p];
  const float invi = ws[OFF_INV_INS + bt * PIX_SCENE + p];

  const float* gt0 = lseg_gt + (size_t)bt * C_SEM * HH * WW + (2 * hp) * WW + 2 * wp;
  float* psrow = &ps[m * PS_STRIDE];
  for (int c = 0; c < C_SEM; ++c) {
    const float* pr = gt0 + (size_t)c * HH * WW;
    v2f r0 = *(const v2f*)pr;
    v2f r1 = *(const v2f*)(pr + WW);
    float v = 0.25f * (r0.x + r0.y + r1.x + r1.y) * invg;
    atomicAdd(&psrow[c], v);
  }
  const float* ri0 = rins + (size_t)bt * C_INS * PIX_SCENE + p;
  float* pirow = &pi[m * PI_STRIDE];
  for (int c = 0; c < C_INS; ++c) atomicAdd(&pirow[c], ri0[c * PIX_SCENE] * invi);
  __syncthreads();

  for (int i = tid; i < NUM_IDS * C_SEM; i += blockDim.x) {
    int mm = i / C_SEM, cc = i - mm * C_SEM;
    atomicAdd(&ws[OFF_PS_SUM + (size_t)bt * NUM_IDS * C_SEM + i], ps[mm * PS_STRIDE + cc]);
  }
  for (int i = tid; i < NUM_IDS * C_INS; i += blockDim.x) {
    int mm = i / C_INS, cc = i - mm * C_INS;
    atomicAdd(&ws[OFF_PI_SUM + (size_t)bt * NUM_IDS * C_INS + i], pi[mm * PI_STRIDE + cc]);
  }
}

// ---------------- K1c: normalize prototypes in place (one wave per segment) -----
__global__ void k1c_normalize(float* __restrict__ ws) {
  const int seg  = blockIdx.x;      // 0..199
  const int lane = threadIdx.x;     // 0..31
  float* psum = ws + OFF_PS_SUM + (size_t)seg * C_SEM;
  float vals[16];
  float ss = 0.0f;
  for (int j = 0; j < 16; ++j) { vals[j] = psum[lane + j * 32]; ss += vals[j] * vals[j]; }
  ss = wave_sum(ss);
  float inv = 1.0f / fmaxf(sqrtf(ss), 1e-12f);
  for (int j = 0; j < 16; ++j) psum[lane + j * 32] = vals[j] * inv;

  float* pisum = ws + OFF_PI_SUM + (size_t)seg * C_INS;
  float a = pisum[lane], b = pisum[lane + 32];
  float s2 = wave_sum(a * a + b * b);
  float inv2 = 1.0f / fmaxf(sqrtf(s2), 1e-12f);
  pisum[lane] = a * inv2;
  pisum[lane + 32] = b * inv2;
}

// ---------------- K2: single-pass per-pixel losses ------------------------------
__global__ void k2_pixel_losses(const float* __restrict__ rsem,
                                const float* __restrict__ msem,
                                const float* __restrict__ rins,
                                const float* __restrict__ mins,
                                const float* __restrict__ mmask,
                                const int*   __restrict__ imask,
                                float* __restrict__ ws) {
  __shared__ float ps[NUM_IDS * PS_STRIDE];
  __shared__ float pi[NUM_IDS * PI_STRIDE];
  __shared__ float cs[NUM_IDS];
  const int tid  = threadIdx.x;
  const int bt   = blockIdx.x / TILES_SCENE;
  const int tile = blockIdx.x % TILES_SCENE;

  for (int i = tid; i < NUM_IDS * C_SEM; i += blockDim.x) {
    int mm = i / C_SEM, cc = i - mm * C_SEM;
    ps[mm * PS_STRIDE + cc] = ws[OFF_PS_SUM + (size_t)bt * NUM_IDS * C_SEM + i];
  }
  for (int i = tid; i < NUM_IDS * C_INS; i += blockDim.x) {
    int mm = i / C_INS, cc = i - mm * C_INS;
    pi[mm * PI_STRIDE + cc] = ws[OFF_PI_SUM + (size_t)bt * NUM_IDS * C_INS + i];
  }
  if (tid < NUM_IDS) cs[tid] = ws[OFF_CNT + bt * NUM_IDS + tid];
  __syncthreads();

  const int p  = tile * TILE_PIX + tid;
  const int hp = p >> 6, wp = p & 63;
  const int m  = imask[(size_t)bt * HH * WW + (2 * hp) * WW + 2 * wp];

  const float* rs0 = rsem + (size_t)bt * C_SEM * PIX_SCENE + p;
  const float* ms0 = msem + (size_t)bt * C_SEM * PIX_SCENE + p;
  const float* prow = &ps[m * PS_STRIDE];
  float nps = 0.f, nms = 0.f, d1 = 0.f, d2 = 0.f;
  for (int c = 0; c < C_SEM; ++c) {
    float a = rs0[(size_t)c * PIX_SCENE];
    float b = ms0[(size_t)c * PIX_SCENE];
    float pr = prow[c];
    nps += a * a; nms += b * b; d1 += a * pr; d2 += a * b;
  }
  const float* ri0 = rins + (size_t)bt * C_INS * PIX_SCENE + p;
  const float* mi0 = mins + (size_t)bt * C_INS * PIX_SCENE + p;
  const float* irow = &pi[m * PI_STRIDE];
  float npi = 0.f, nmi = 0.f, d3 = 0.f, d4 = 0.f;
  for (int c = 0; c < C_INS; ++c) {
    float a = ri0[c * PIX_SCENE];
    float b = mi0[c * PIX_SCENE];
    float pr = irow[c];
    npi += a * a; nmi += b * b; d3 += a * pr; d4 += a * b;
  }
  const float mm_v = mmask[(size_t)bt * PIX_SCENE + p];
  const float ips = 1.0f / fmaxf(sqrtf(nps), 1e-12f);
  const float ims = 1.0f / fmaxf(sqrtf(nms), 1e-12f);
  const float ipi = 1.0f / fmaxf(sqrtf(npi), 1e-12f);
  const float imi = 1.0f / fmaxf(sqrtf(nmi), 1e-12f);
  const float valid = (m > 0 && cs[m] >= 2.0f) ? 1.0f : 0.0f;

  float sa = valid * (1.0f - d1 * ips);
  float sm = mm_v * (1.0f - d2 * ips * ims);
  float ii = valid * (1.0f - d3 * ipi);
  float im = mm_v * (1.0f - d4 * ipi * imi);

  sa = wave_sum(sa); float vc = wave_sum(valid); sm = wave_sum(sm);
  float ms_s = wave_sum(mm_v); ii = wave_sum(ii); im = wave_sum(im);
  if ((tid & 31) == 0) {
    atomicAdd(&ws[OFF_ACC + 0], sa);
    atomicAdd(&ws[OFF_ACC + 1], vc);
    atomicAdd(&ws[OFF_ACC + 2], sm);
    atomicAdd(&ws[OFF_ACC + 3], ms_s);
    atomicAdd(&ws[OFF_ACC + 4], ii);
    atomicAdd(&ws[OFF_ACC + 5], im);
  }
}

// ---------------- K3: inter-instance hinge via f32 WMMA Gram matrix -------------
__global__ void k3_inter_wmma(float* __restrict__ ws) {
  __shared__ float P[32 * 64];   // zero-padded prototypes, row-major [32][64]
  __shared__ float vm[32];
  const int bt = blockIdx.x;     // 0..7
  const int lane = threadIdx.x;  // 0..31
  for (int i = lane; i < NUM_IDS * C_INS; i += 32)
    P[i] = ws[OFF_PI_SUM + (size_t)bt * NUM_IDS * C_INS + i];
  for (int i = NUM_IDS * C_INS + lane; i < 32 * 64; i += 32) P[i] = 0.0f;
  {
    float c = (lane >= 1 && lane < NUM_IDS) ? ws[OFF_CNT + bt * NUM_IDS + lane] : 0.0f;
    vm[lane] = (lane >= 1 && lane < NUM_IDS && c >= 2.0f) ? 1.0f : 0.0f;
  }
  __syncthreads();

  const int r = lane & 15;
  const int half = lane >> 4;
  v8f c00 = {0.f,0.f,0.f,0.f,0.f,0.f,0.f,0.f};
  v8f c01 = {0.f,0.f,0.f,0.f,0.f,0.f,0.f,0.f};
  v8f c11 = {0.f,0.f,0.f,0.f,0.f,0.f,0.f,0.f};
  // Gram = P * P^T ; A(16x4) and B(4x16) fragments coincide for the mirror layouts
  for (int kk = 0; kk < 16; ++kk) {
    const int k = kk * 4 + half * 2;
    v2f a0 = *(const v2f*)&P[r * 64 + k];
    v2f a1 = *(const v2f*)&P[(16 + r) * 64 + k];
    c00 = __builtin_amdgcn_wmma_f32_16x16x4_f32(false, a0, false, a0, (short)0, c00, false, false);
    c01 = __builtin_amdgcn_wmma_f32_16x16x4_f32(false, a0, false, a1, (short)0, c01, false, false);
    c11 = __builtin_amdgcn_wmma_f32_16x16x4_f32(false, a1, false, a1, (short)0, c11, false, false);
  }
  // C/D layout: VGPR i -> M = i + 8*half, N = lane&15
  float hs = 0.0f, pc = 0.0f;
  const int n_loc = lane & 15;
  const int mbase = half * 8;
#pragma unroll
  for (int i = 0; i < 8; ++i) {
    const int ml = mbase + i;
    float neq = (ml != n_loc) ? 1.0f : 0.0f;
    float pm0 = vm[ml] * vm[n_loc] * neq;                  // tile (0,0): ids 0..15
    hs += pm0 * fmaxf(c00[i] - 0.2f, 0.0f); pc += pm0;
    float pm1 = vm[16 + ml] * vm[16 + n_loc] * neq;        // tile (1,1): ids 16..31
    hs += pm1 * fmaxf(c11[i] - 0.2f, 0.0f); pc += pm1;
    float pm2 = vm[ml] * vm[16 + n_loc] * 2.0f;            // tile (0,1) + mirror
    hs += pm2 * fmaxf(c01[i] - 0.2f, 0.0f); pc += pm2;
  }
  hs = wave_sum(hs); pc = wave_sum(pc);
  if (lane == 0) {
    atomicAdd(&ws[OFF_ACC + 6], hs);
    atomicAdd(&ws[OFF_ACC + 7], pc);
  }
}

// ---------------- K4: combine ---------------------------------------------------
__global__ void k4_final(const float* __restrict__ ws, float* __restrict__ out) {
  if (threadIdx.x == 0 && blockIdx.x == 0) {
    float sa = ws[OFF_ACC + 0] / fmaxf(ws[OFF_ACC + 1], 1.0f);
    float sm = ws[OFF_ACC + 2] / fmaxf(ws[OFF_ACC + 3], 1.0f);
    float ii = ws[OFF_ACC + 4] / fmaxf(ws[OFF_ACC + 1], 1.0f);
    float im = ws[OFF_ACC + 5] / fmaxf(ws[OFF_ACC + 3], 1.0f);
    float inter = ws[OFF_ACC + 6] / fmaxf(ws[OFF_ACC + 7], 1.0f);
    out[0] = 0.5f * sa + sm + (ii + inter) + im;
  }
}

extern "C" void kernel_launch(void* const* d_in, const int* in_sizes, int n_in,
                              void* d_out, int out_size, void* d_ws, size_t ws_size,
                              hipStream_t stream) {
  const float* rsem  = (const float*)d_in[0];
  const float* rins  = (const float*)d_in[1];
  const float* gt    = (const float*)d_in[2];
  const float* msem  = (const float*)d_in[3];
  const float* mins  = (const float*)d_in[4];
  const float* mmask = (const float*)d_in[5];
  const int*   im    = (const int*)d_in[6];
  float* ws = (float*)d_ws;

  // zero proto sums + counts + accumulators (graph-capture safe)
  hipMemsetAsync(ws, 0, (size_t)(OFF_ACC + 8) * sizeof(float), stream);

  k1a_norms_counts<<<N_BT * TILES_SCENE, 256, 0, stream>>>(gt, rins, im, ws);
  k1b_proto_accum <<<N_BT * TILES_SCENE, 256, 0, stream>>>(gt, rins, im, ws);
  k1c_normalize   <<<N_BT * NUM_IDS, 32, 0, stream>>>(ws);
  k2_pixel_losses <<<N_BT * TILES_SCENE, 256, 0, stream>>>(rsem, msem, rins, mins, mmask, im, ws);
  k3_inter_wmma   <<<N_BT, 32, 0, stream>>>(ws);
  k4_final        <<<1, 1, 0, stream>>>(ws, (float*)d_out);
}